// DISTANA_4406636445738
// MI455X (gfx1250) — compile-verified
//
#include <hip/hip_runtime.h>

typedef __attribute__((ext_vector_type(2))) float v2f;
typedef __attribute__((ext_vector_type(8))) float v8f;

#define B_      128
#define N_      4096
#define TOT     (B_ * N_)   // 524288 nodes
#define IN_     9
#define PRE_    4
#define CELLS_  16
#define OUT_    9

#define WAVES_PER_BLOCK 8
#define LDS_STRIDE      18   // dwords/row: 72B -> 8B aligned for b64, avoids bank conflicts

// Branch-free transcendentals (straight-line VALU: v_exp_f32 + v_rcp_f32).
// tanh(x) = 1 - 2/(exp(2x)+1): saturates to +/-1, no inf*0 NaN path.
__device__ __forceinline__ float fast_tanh(float x) {
    const float e = __expf(2.0f * x);
    return 1.0f - 2.0f * __builtin_amdgcn_rcpf(e + 1.0f);
}
__device__ __forceinline__ float fast_sigmoid(float x) {
    return __builtin_amdgcn_rcpf(1.0f + __expf(-x));
}

// DISTANA fused step, 16 nodes per wave32.
//   stage1: pre = tanh(x @ Wpre)        per-lane dual dot-9, lands in WMMA A-layout
//   stage2: net = pre @ Wlstm + old_h   exact V_WMMA_F32_16X16X4_F32 (M16,N16,K4)
//   CIFG:   i=sig(net), g=tanh(net), nc=(1-i)*oc+i*g, nh=tanh(nc)   (D-layout, coalesced)
//   stage3: y = tanh(nh @ Wpost)        4 chained K=4 WMMAs via LDS transpose
__global__ __launch_bounds__(256) void distana_step_kernel(
    const float* __restrict__ x,        // (TOT, 9)
    const float* __restrict__ Wpre,     // (9, 4)
    const float* __restrict__ Wlstm,    // (4, 16)
    const float* __restrict__ Wpost,    // (16, 9)
    const float* __restrict__ old_h,    // (TOT, 16)
    const float* __restrict__ old_cell, // (TOT, 16)
    float* __restrict__ y_out,          // (TOT, 9)
    float* __restrict__ h_out,          // (TOT, 16)
    float* __restrict__ c_out)          // (TOT, 16)
{
    __shared__ float lds[WAVES_PER_BLOCK * 16 * LDS_STRIDE];

    const int lane = threadIdx.x & 31;
    const int wave = threadIdx.x >> 5;
    const int tile = blockIdx.x * WAVES_PER_BLOCK + wave;
    const long long node0 = (long long)tile * 16;

    const int m  = lane & 15;   // node row within tile (A-side) / column index (B,D-side)
    const int hi = lane >> 4;   // 0 = low K/row half, 1 = high half
    float* wlds = &lds[wave * 16 * LDS_STRIDE];

    // ---- Stage 1: pre = tanh(x @ Wpre) -------------------------------------
    // A-layout of V_WMMA_F32_16X16X4_F32 (32-bit A, 16x4):
    //   lanes 0-15  hold {K=0, K=1} of row M=lane
    //   lanes 16-31 hold {K=2, K=3} of row M=lane-16
    const int p0 = hi ? 2 : 0;
    const float* __restrict__ xp = x + (node0 + m) * IN_;
    float acc0 = 0.0f, acc1 = 0.0f;
    #pragma unroll
    for (int i = 0; i < IN_; ++i) {
        const float xi = xp[i];
        acc0 = fmaf(xi, Wpre[i * PRE_ + p0],     acc0);
        acc1 = fmaf(xi, Wpre[i * PRE_ + p0 + 1], acc1);
    }
    v2f a;
    a.x = fast_tanh(acc0);
    a.y = fast_tanh(acc1);

    // ---- Stage 2: net = pre @ Wlstm  (perfect-fit 16x16x4 WMMA) ------------
    // B-layout (4x16): VGPR0 = rows {K0 | K2}, VGPR1 = rows {K1 | K3}, col = lane%16
    v2f b;
    b.x = Wlstm[(hi ? 2 : 0) * CELLS_ + m];
    b.y = Wlstm[(hi ? 3 : 1) * CELLS_ + m];
    v8f zero = {};
    v8f net = __builtin_amdgcn_wmma_f32_16x16x4_f32(false, a, false, b,
                                                    (short)0, zero, false, false);

    // ---- CIFG cell update (elementwise on D-layout) ------------------------
    // D-layout: VGPR r -> row (r + 8*hi), col = lane%16  => 2x 64B coalesced rows.
    // old_h/old_cell/outputs are touched exactly once -> nontemporal.
    #pragma unroll
    for (int r = 0; r < 8; ++r) {
        const long long node = node0 + r + 8 * hi;
        const long long idx  = node * CELLS_ + m;
        const float nv = net[r] + __builtin_nontemporal_load(&old_h[idx]);
        const float oc = __builtin_nontemporal_load(&old_cell[idx]);
        const float ig = fast_sigmoid(nv);
        const float g  = fast_tanh(nv);
        const float nc = (1.0f - ig) * oc + ig * g;
        const float nh = fast_tanh(nc);
        __builtin_nontemporal_store(nc, &c_out[idx]);
        __builtin_nontemporal_store(nh, &h_out[idx]);
        wlds[(r + 8 * hi) * LDS_STRIDE + m] = nh;       // stage for D->A transpose
    }

    // Wave-local: all 32 lanes run in lockstep, just drain the DS counter so
    // cross-lane LDS reads below see the stores above.
    asm volatile("s_wait_dscnt 0" ::: "memory");

    // ---- Stage 3: y = tanh(new_h @ Wpost), K=16 as 4 chained WMMAs ---------
    v8f y = {};
    #pragma unroll
    for (int k = 0; k < 4; ++k) {
        const int kb = 4 * k + 2 * hi;                  // K-base this lane-half supplies
        v2f a3, b3;
        a3.x = wlds[m * LDS_STRIDE + kb];               // new_h[m][kb], new_h[m][kb+1]
        a3.y = wlds[m * LDS_STRIDE + kb + 1];           // (8B-aligned pair -> ds_load_b64)
        b3.x = (m < OUT_) ? Wpost[kb * OUT_ + m]       : 0.0f;  // N padded 9 -> 16
        b3.y = (m < OUT_) ? Wpost[(kb + 1) * OUT_ + m] : 0.0f;
        y = __builtin_amdgcn_wmma_f32_16x16x4_f32(false, a3, false, b3,
                                                  (short)0, y, false, false);
    }
    #pragma unroll
    for (int r = 0; r < 8; ++r) {
        if (m < OUT_) {
            const long long node = node0 + r + 8 * hi;
            __builtin_nontemporal_store(fast_tanh(y[r]), &y_out[node * OUT_ + m]);
        }
    }
}

extern "C" void kernel_launch(void* const* d_in, const int* in_sizes, int n_in,
                              void* d_out, int out_size, void* d_ws, size_t ws_size,
                              hipStream_t stream) {
    const float* x     = (const float*)d_in[0];   // input  (B,N,9,1)
    const float* Wpre  = (const float*)d_in[1];   // (9,4)
    const float* Wlstm = (const float*)d_in[2];   // (4,16)
    const float* Wpost = (const float*)d_in[3];   // (16,9)
    const float* oh    = (const float*)d_in[4];   // (B,N,16)
    const float* oc    = (const float*)d_in[5];   // (B,N,16)

    float* out   = (float*)d_out;                 // y | new_h | new_cell, flat
    float* y_out = out;
    float* h_out = y_out + (size_t)TOT * OUT_;
    float* c_out = h_out + (size_t)TOT * CELLS_;

    const int tiles  = TOT / 16;                  // 32768 waves
    const int blocks = tiles / WAVES_PER_BLOCK;   // 4096 blocks of 256 threads
    distana_step_kernel<<<blocks, 256, 0, stream>>>(x, Wpre, Wlstm, Wpost, oh, oc,
                                                    y_out, h_out, c_out);
}